// ConstantODEblock_57535381897864
// MI455X (gfx1250) — compile-verified
//
#include <hip/hip_runtime.h>
#include <math.h>

#define D_FEAT 128
#define WPB    8     // waves per block (wave32)
#define TPB    256

#if defined(__gfx1250__) && __has_builtin(__builtin_amdgcn_global_load_async_to_lds_b32) && __has_builtin(__builtin_amdgcn_s_wait_asynccnt)
#define HAVE_ASYNC 1
#else
#define HAVE_ASYNC 0
#endif

#if defined(__AMDGCN__) && !HAVE_ASYNC
#warning "gfx1250 async-to-LDS builtin not available on device pass: using __shfl fallback"
#endif

#if HAVE_ASYNC
typedef __attribute__((address_space(1))) int GInt;   // global
typedef __attribute__((address_space(3))) int LInt;   // LDS
#endif

// Guaranteed single-instruction L2 atomic (no CAS-loop expansion).
// No-return form -> tracked by STOREcnt; s_endpgm waits idle implicitly.
__device__ __forceinline__ void atom_add_f32(float* p, float v) {
  asm volatile("global_atomic_add_f32 %0, %1, off scope:SCOPE_DEV"
               : : "v"(p), "v"(v));
}

// ---------------------------------------------------------------- zero ----
__global__ __launch_bounds__(TPB) void grand_zero_kernel(float4* p, int n4) {
  int i = blockIdx.x * blockDim.x + threadIdx.x;
  int stride = gridDim.x * blockDim.x;
  float4 z; z.x = z.y = z.z = z.w = 0.0f;
  for (; i < n4; i += stride) p[i] = z;
}

// ------------------------------------------------------------- scatter ----
// acc[row[e], :] += w[e] * h[col[e], :]   (one wave per edge, float4/lane)
__global__ __launch_bounds__(TPB) void grand_scatter_kernel(
    const float* __restrict__ h, const int* __restrict__ row,
    const int* __restrict__ col, const float* __restrict__ ew,
    float* __restrict__ acc, int E)
{
  const int lane    = threadIdx.x & 31;
  const int wv      = threadIdx.x >> 5;
  const int nbatch  = (E + 31) >> 5;          // 32 edges per batch
  const int wstride = gridDim.x * WPB;
  const size_t lo   = (size_t)(lane << 2);    // lane's 4-float slice offset
  int b = blockIdx.x * WPB + wv;
  if (b >= nbatch) return;

#if HAVE_ASYNC
  __shared__ int   s_row[2][WPB][32];
  __shared__ int   s_col[2][WPB][32];
  __shared__ float s_wgt[2][WPB][32];

  int cur = 0;
  { // stage first batch (async DMA to LDS, per-lane b32)
    int e = b * 32 + lane; if (e >= E) e = E - 1;
    __builtin_amdgcn_global_load_async_to_lds_b32((GInt*)(row + e), (LInt*)&s_row[cur][wv][lane], 0, 0);
    __builtin_amdgcn_global_load_async_to_lds_b32((GInt*)(col + e), (LInt*)&s_col[cur][wv][lane], 0, 0);
    __builtin_amdgcn_global_load_async_to_lds_b32((GInt*)(ew  + e), (LInt*)&s_wgt[cur][wv][lane], 0, 0);
  }
  __builtin_amdgcn_s_wait_asynccnt(0);

  for (; b < nbatch; b += wstride) {
    const int nb = b + wstride;
    if (nb < nbatch) {  // overlap staging of next batch with compute
      int e = nb * 32 + lane; if (e >= E) e = E - 1;
      int nxt = cur ^ 1;
      __builtin_amdgcn_global_load_async_to_lds_b32((GInt*)(row + e), (LInt*)&s_row[nxt][wv][lane], 0, 0);
      __builtin_amdgcn_global_load_async_to_lds_b32((GInt*)(col + e), (LInt*)&s_col[nxt][wv][lane], 0, 0);
      __builtin_amdgcn_global_load_async_to_lds_b32((GInt*)(ew  + e), (LInt*)&s_wgt[nxt][wv][lane], 0, 0);
    }
    int cnt = E - b * 32; if (cnt > 32) cnt = 32;
    {
      // software pipeline: load h-row for edge j+1 before edge j's atomics
      int c0 = s_col[cur][wv][0];
      float4 hv = *(const float4*)(h + (((size_t)c0) << 7) + lo);
      for (int j = 0; j < cnt; ++j) {
        const int    r  = s_row[cur][wv][j];   // LDS broadcast reads
        const float  wt = s_wgt[cur][wv][j];
        const float4 cv = hv;
        if (j + 1 < cnt) {
          const int cn = s_col[cur][wv][j + 1];
          hv = *(const float4*)(h + (((size_t)cn) << 7) + lo);
        }
        float* ap = acc + (((size_t)r) << 7) + lo;
        atom_add_f32(ap + 0, wt * cv.x);
        atom_add_f32(ap + 1, wt * cv.y);
        atom_add_f32(ap + 2, wt * cv.z);
        atom_add_f32(ap + 3, wt * cv.w);
      }
    }
    __builtin_amdgcn_s_wait_asynccnt(0);
    cur ^= 1;
  }
#else
  // Fallback: per-lane register staging + wave shuffles
  for (; b < nbatch; b += wstride) {
    const int base = b * 32;
    int cnt = E - base; if (cnt > 32) cnt = 32;
    int e = base + lane; if (e >= E) e = E - 1;
    const int   r0 = row[e];
    const int   c0 = col[e];
    const float w0 = ew[e];
    for (int j = 0; j < cnt; ++j) {
      const int   r  = __shfl(r0, j, 32);
      const int   c  = __shfl(c0, j, 32);
      const float wt = __shfl(w0, j, 32);
      const float4 hv = *(const float4*)(h + (((size_t)c) << 7) + lo);
      float* ap = acc + (((size_t)r) << 7) + lo;
      atom_add_f32(ap + 0, wt * hv.x);
      atom_add_f32(ap + 1, wt * hv.y);
      atom_add_f32(ap + 2, wt * hv.z);
      atom_add_f32(ap + 3, wt * hv.w);
    }
  }
#endif
}

// -------------------------------------------------------------- update ----
// out = h + alpha * (acc - h), alpha = sigmoid(alpha_raw[0]).
// NOTE: out may alias acc (in-place per element) -> no __restrict__ here.
__global__ __launch_bounds__(TPB) void grand_update_kernel(
    const float4* h, const float4* acc, const float* alpha_raw,
    float4* out, int n4)
{
  const float a     = alpha_raw[0];
  const float alpha = 1.0f / (1.0f + __expf(-a));
  int i = blockIdx.x * blockDim.x + threadIdx.x;
  int stride = gridDim.x * blockDim.x;
  for (; i < n4; i += stride) {
    const float4 hv = h[i];
    const float4 av = acc[i];
    float4 o;
    o.x = fmaf(alpha, av.x - hv.x, hv.x);
    o.y = fmaf(alpha, av.y - hv.y, hv.y);
    o.z = fmaf(alpha, av.z - hv.z, hv.z);
    o.w = fmaf(alpha, av.w - hv.w, hv.w);
    out[i] = o;
  }
}

// ---------------------------------------------------------------- host ----
extern "C" void kernel_launch(void* const* d_in, const int* in_sizes, int n_in,
                              void* d_out, int out_size, void* d_ws, size_t ws_size,
                              hipStream_t stream) {
  const float* x     = (const float*)d_in[0];   // [N,128] f32
  const float* ew    = (const float*)d_in[1];   // [E]     f32
  const float* alpha = (const float*)d_in[2];   // [1]     f32 (pre-sigmoid)
  const int*   ei    = (const int*)  d_in[3];   // [2,E]   i32: row then col

  const int E = in_sizes[1];
  const int N = in_sizes[0] / D_FEAT;
  const int n4 = (N * D_FEAT) / 4;

  const int* row = ei;
  const int* col = ei + E;

  float* w0 = (float*)d_ws;                     // [N,128]
  float* w1 = w0 + (size_t)N * D_FEAT;          // [N,128]
  float* out = (float*)d_out;

  const int gElem    = (n4 + TPB - 1) / TPB;    // elementwise grids
  const int gScatter = 1024;                    // 8192 waves, ~3 batches/wave

  const float* h = x;
  for (int s = 0; s < 4; ++s) {
    float* acc = (s & 1) ? w1 : w0;
    grand_zero_kernel<<<gElem, TPB, 0, stream>>>((float4*)acc, n4);
    grand_scatter_kernel<<<gScatter, TPB, 0, stream>>>(h, row, col, ew, acc, E);
    float* nxt = (s == 3) ? out : acc;          // in-place into acc buffer
    grand_update_kernel<<<gElem, TPB, 0, stream>>>(
        (const float4*)h, (const float4*)acc, alpha, (float4*)nxt, n4);
    h = nxt;
  }
}